// UniformViewTransformer_68831145886085
// MI455X (gfx1250) — compile-verified
//
#include <hip/hip_runtime.h>

#define NCAMS   6
#define C_FEAT  64
#define HF      16
#define WF      44
#define GX      128
#define GY      128
#define GZ      8
#define P_TOT   (GX * GY * GZ)          // 131072 voxels
#define FEAT_N  (NCAMS * C_FEAT * HF * WF) // 270336 floats
#define IMG_H   256
#define IMG_W   704

typedef unsigned int u32x4 __attribute__((ext_vector_type(4)));
typedef int          i32x8 __attribute__((ext_vector_type(8)));
typedef int          i32x4 __attribute__((ext_vector_type(4)));
typedef float        v2f   __attribute__((ext_vector_type(2)));
typedef float        v8f   __attribute__((ext_vector_type(8)));

__device__ __forceinline__ int clampi(int v, int lo, int hi) {
    return v < lo ? lo : (v > hi ? hi : v);
}

// ---------------------------------------------------------------------------
// Kernel 1: proj[n] = cam2img[n] @ ego2cam[n]  (6 independent 4x4 matmuls)
// done on the matrix unit via V_WMMA_F32_16X16X4_F32.
//   WMMA #1: A = rows of cam2img[0..3] stacked (16x4), B = cols of
//            ego2cam[0..3] (4x16); diagonal 4x4 blocks of D = proj[0..3].
//   WMMA #2: same trick for cams 4,5 in block slots 0,1 (rest zero).
// Single full wave32 so EXEC is all-ones at the WMMA (ISA requirement).
// ---------------------------------------------------------------------------
__global__ void prep_proj_wmma(const float* __restrict__ ego2cam,
                               const float* __restrict__ cam2img,
                               float* __restrict__ proj) {
    int lane = threadIdx.x;      // 0..31, one wave
    int h    = lane >> 4;        // 0: K=0,1   1: K=2,3
    int m    = lane & 15;        // A row / B col index
    int cA   = m >> 2, iA = m & 3;   // camera block, row-within-block
    int jB   = m & 3;                // col-within-block (B col = m)

    // --- WMMA #1 operands: cams 0..3 ---
    v2f a1, b1;
    a1.x = cam2img[cA * 16 + iA * 4 + (2 * h + 0)];
    a1.y = cam2img[cA * 16 + iA * 4 + (2 * h + 1)];
    b1.x = ego2cam[cA * 16 + (2 * h + 0) * 4 + jB];
    b1.y = ego2cam[cA * 16 + (2 * h + 1) * 4 + jB];

    // --- WMMA #2 operands: cams 4,5 in block slots 0,1; slots 2,3 zero ---
    v2f a2 = { 0.f, 0.f }, b2 = { 0.f, 0.f };
    if (m < 8) {
        int c2 = 4 + (m >> 2);
        a2.x = cam2img[c2 * 16 + (m & 3) * 4 + (2 * h + 0)];
        a2.y = cam2img[c2 * 16 + (m & 3) * 4 + (2 * h + 1)];
        b2.x = ego2cam[c2 * 16 + (2 * h + 0) * 4 + (m & 3)];
        b2.y = ego2cam[c2 * 16 + (2 * h + 1) * 4 + (m & 3)];
    }

    v8f c0 = { 0.f, 0.f, 0.f, 0.f, 0.f, 0.f, 0.f, 0.f };
    // 8 args: (neg_a, A, neg_b, B, c_mod, C, reuse_a, reuse_b)
    v8f d1 = __builtin_amdgcn_wmma_f32_16x16x4_f32(false, a1, false, b1,
                                                   (short)0, c0, false, false);
    v8f d2 = __builtin_amdgcn_wmma_f32_16x16x4_f32(false, a2, false, b2,
                                                   (short)0, c0, false, false);

    // D layout: lanes 0-15 -> VGPR r = D[r][lane]; lanes 16-31 -> D[r+8][lane-16]
    // Diagonal block of D#1 for this lane's column (camera cA) lives in
    // rows M = 4*cA .. 4*cA+3, which are in half (cA>>1).
    if ((cA >> 1) == h) {
        #pragma unroll
        for (int i = 0; i < 4; ++i) {
            int M = 4 * cA + i;
            proj[cA * 16 + i * 4 + jB] = d1[M - 8 * h];
        }
    }
    // D#2 diagonal blocks 0,1 -> cams 4,5 (rows 0..7 => half 0 only)
    if (h == 0 && m < 8) {
        int c2 = m >> 2;   // 0 -> cam4, 1 -> cam5
        #pragma unroll
        for (int i = 0; i < 4; ++i)
            proj[(4 + c2) * 16 + i * 4 + (m & 3)] = d2[4 * c2 + i];
    }
}

// ---------------------------------------------------------------------------
// Kernel 2: (N,C,H,W) -> (N,H,W,C) so channels are contiguous per pixel.
// Exactly FEAT_N = 1056*256 threads; coalesced writes, L2-absorbed reads.
// ---------------------------------------------------------------------------
__global__ void chan_last(const float* __restrict__ in, float* __restrict__ outp) {
    int gid = blockIdx.x * 256 + threadIdx.x;
    int c = gid & (C_FEAT - 1);
    int w = (gid >> 6) % WF;
    int t = gid / (C_FEAT * WF);
    int h = t & (HF - 1);
    int n = t >> 4;
    outp[gid] = in[((n * C_FEAT + c) * HF + h) * WF + w];
}

// ---------------------------------------------------------------------------
// Kernel 3: per-voxel projection + bilinear sample + camera-average fusion.
// Projection matrices staged into LDS via the Tensor Data Mover (TDM),
// synchronized with s_wait_tensorcnt (gfx1250 async-tensor path).
// ---------------------------------------------------------------------------
__global__ void fuse(const float* __restrict__ feats,   // (N,H,W,C) channel-last
                     const float* __restrict__ gproj,   // 96 floats in global
                     const float* __restrict__ vcoords, // (3, P)
                     float* __restrict__ outp) {        // (C, P)
    __shared__ float sproj[128]; // 96 used, padded

    // --- TDM stage: global -> LDS, 96 f32 as a 96x1 tile (D# built inline) ---
    if (threadIdx.x < 32) {
        unsigned long long ga = (unsigned long long)gproj;
        unsigned int lds_off  = (unsigned int)(unsigned long long)&sproj[0];
        // D# group0: count=1(valid), lds_addr, global_addr[56:0], type=2
        u32x4 g0 = { 1u, lds_off, (unsigned int)ga,
                     (unsigned int)(ga >> 32) | (2u << 30) };
        // D# group1: data_size=2 (4B); tensor_dim0=96; tensor_dim1=1;
        //            tile_dim0=96; tile_dim1=1; dim0_stride=96
        i32x8 g1 = { (int)(2u << 16),        // workgroup_mask=0, data_size=4B
                     (int)(96u << 16),       // tensor_dim0[15:0]
                     (int)(1u  << 16),       // tensor_dim0 hi=0 | tensor_dim1 lo=1
                     (int)(96u << 16),       // tensor_dim1 hi=0 | tile_dim0=96
                     1,                      // tile_dim1=1, tile_dim2=0
                     96,                     // tensor_dim0_stride lo
                     0, 0 };                 // stride hi, dim1_stride=0
        i32x4 gz4 = { 0, 0, 0, 0 };
        i32x8 gz8 = { 0, 0, 0, 0, 0, 0, 0, 0 };
        __builtin_amdgcn_tensor_load_to_lds(g0, g1, gz4, gz4, gz8, 0);
        __builtin_amdgcn_s_wait_tensorcnt(0);
    }
    __syncthreads();

    int pid = blockIdx.x * blockDim.x + threadIdx.x; // 0 .. P_TOT-1
    float vx = vcoords[pid];
    float vy = vcoords[P_TOT + pid];
    float vz = vcoords[2 * P_TOT + pid];

    float4 acc[16];
    #pragma unroll
    for (int k = 0; k < 16; ++k) acc[k] = make_float4(0.f, 0.f, 0.f, 0.f);
    float cnt = 0.f;

    #pragma unroll 1
    for (int cam = 0; cam < NCAMS; ++cam) {
        const float* M = &sproj[cam * 16];
        float Xp = M[0] * vx + M[1] * vy + M[2]  * vz + M[3];
        float Yp = M[4] * vx + M[5] * vy + M[6]  * vz + M[7];
        float Dp = M[8] * vx + M[9] * vy + M[10] * vz + M[11];

        float inv = 1.0f / (Dp + 1e-6f);
        float u = Xp * inv;
        float v = Yp * inv;
        bool valid = (Dp > 0.1f) &&
                     (u >= 0.f) && (u <= (float)(IMG_W - 1)) &&
                     (v >= 0.f) && (v <= (float)(IMG_H - 1));
        if (!valid) continue;
        cnt += 1.f;

        // align_corners sampling coordinate collapses to u/STRIDE, v/STRIDE
        float xf = u * 0.0625f;
        float yf = v * 0.0625f;
        float x0 = floorf(xf), y0 = floorf(yf);
        float wx1 = xf - x0, wx0 = 1.f - wx1;
        float wy1 = yf - y0, wy0 = 1.f - wy1;
        int x0i = (int)x0, y0i = (int)y0;
        int x1i = x0i + 1, y1i = y0i + 1;

        float mx0 = (x0i >= 0 && x0i < WF) ? 1.f : 0.f;
        float mx1 = (x1i >= 0 && x1i < WF) ? 1.f : 0.f;
        float my0 = (y0i >= 0 && y0i < HF) ? 1.f : 0.f;
        float my1 = (y1i >= 0 && y1i < HF) ? 1.f : 0.f;

        int x0c = clampi(x0i, 0, WF - 1), x1c = clampi(x1i, 0, WF - 1);
        int y0c = clampi(y0i, 0, HF - 1), y1c = clampi(y1i, 0, HF - 1);

        float w00 = wx0 * wy0 * (mx0 * my0);
        float w01 = wx1 * wy0 * (mx1 * my0);
        float w10 = wx0 * wy1 * (mx0 * my1);
        float w11 = wx1 * wy1 * (mx1 * my1);

        const float* base = feats + cam * (HF * WF * C_FEAT);
        const float4* p00 = (const float4*)(base + (y0c * WF + x0c) * C_FEAT);
        const float4* p01 = (const float4*)(base + (y0c * WF + x1c) * C_FEAT);
        const float4* p10 = (const float4*)(base + (y1c * WF + x0c) * C_FEAT);
        const float4* p11 = (const float4*)(base + (y1c * WF + x1c) * C_FEAT);

        #pragma unroll
        for (int k = 0; k < 16; ++k) {
            float4 f00 = p00[k], f01 = p01[k], f10 = p10[k], f11 = p11[k];
            acc[k].x += w00 * f00.x + w01 * f01.x + w10 * f10.x + w11 * f11.x;
            acc[k].y += w00 * f00.y + w01 * f01.y + w10 * f10.y + w11 * f11.y;
            acc[k].z += w00 * f00.z + w01 * f01.z + w10 * f10.z + w11 * f11.z;
            acc[k].w += w00 * f00.w + w01 * f01.w + w10 * f10.w + w11 * f11.w;
        }
    }

    float scale = 1.0f / fmaxf(cnt, 1.0f);
    // out[c*P + pid]: lanes hold consecutive pid -> fully coalesced stores
    #pragma unroll
    for (int k = 0; k < 16; ++k) {
        outp[(4 * k + 0) * P_TOT + pid] = acc[k].x * scale;
        outp[(4 * k + 1) * P_TOT + pid] = acc[k].y * scale;
        outp[(4 * k + 2) * P_TOT + pid] = acc[k].z * scale;
        outp[(4 * k + 3) * P_TOT + pid] = acc[k].w * scale;
    }
}

// ---------------------------------------------------------------------------
extern "C" void kernel_launch(void* const* d_in, const int* in_sizes, int n_in,
                              void* d_out, int out_size, void* d_ws, size_t ws_size,
                              hipStream_t stream) {
    const float* img_feats = (const float*)d_in[0]; // (6,64,16,44)
    const float* ego2cam   = (const float*)d_in[1]; // (6,4,4)
    const float* cam2img   = (const float*)d_in[2]; // (6,4,4)
    const float* vcoords   = (const float*)d_in[3]; // (3,8,128,128)
    float* out = (float*)d_out;                     // (1,64,8,128,128)

    float* proj  = (float*)d_ws;                      // 96 floats
    float* feats = (float*)((char*)d_ws + 1024);      // 270336 floats (~1.05 MB)

    prep_proj_wmma<<<1, 32, 0, stream>>>(ego2cam, cam2img, proj);
    chan_last<<<FEAT_N / 256, 256, 0, stream>>>(img_feats, feats);
    fuse<<<P_TOT / 256, 256, 0, stream>>>(feats, proj, vcoords, out);
}